// TimeFreqMasking_47897475285313
// MI455X (gfx1250) — compile-verified
//
#include <hip/hip_runtime.h>
#include <hip/hip_bf16.h>

typedef __attribute__((ext_vector_type(2))) float v2f;
typedef __attribute__((ext_vector_type(8))) float v8f;

#define BS 32
#define NP 64
#define NV 321
#define PL 12
#define DM 128
#define KT 32   /* int(64*0.5) patches masked by time */
#define KF 25   /* int(64*0.4) patches masked by freq */

// ws layout (floats): [0..191] M[12][16] (N padded w/ zeros)
//                     [192..207] cvec, [208..223] tvec, [224..239] fvec
__global__ __launch_bounds__(256) void tf_precompute(
    const float* __restrict__ W_in, const float* __restrict__ b_in,
    const float* __restrict__ W_out, const float* __restrict__ b_out,
    const float* __restrict__ tmt, const float* __restrict__ fmt,
    float* __restrict__ ws) {
  const int t = threadIdx.x;
  if (t < 192) {
    const int l = t >> 4, n = t & 15;
    float acc = 0.f;
    if (n < PL) {
      for (int d = 0; d < DM; ++d) acc += W_in[l * DM + d] * W_out[d * PL + n];
    }
    ws[t] = acc;                                   // M[l][n], zero-padded cols 12..15
  } else if (t < 208) {
    const int n = t - 192;
    float val = 0.f;
    if (n < PL) {
      float s = 0.f;
      for (int d = 0; d < DM; ++d) s += b_in[d] * W_out[d * PL + n];
      val = s + b_out[n];
    }
    ws[192 + n] = val;                             // cvec
  } else if (t < 240) {
    const bool isT = (t < 224);
    const int n = (t - 208) & 15;
    float val = 0.f;
    if (n < PL) {
      float cs = 0.f;
      for (int d = 0; d < DM; ++d) cs += W_out[d * PL + n];
      val = (isT ? tmt[0] : fmt[0]) * cs + b_out[n];
    }
    ws[(isT ? 208 : 224) + n] = val;               // tvec / fvec
  }
}

__global__ __launch_bounds__(64) void tf_fused(
    const float* __restrict__ x, const float* __restrict__ ws,
    float* __restrict__ out) {
  __shared__ float x_lds[NP][PL];
  __shared__ float cv_lds[NP];
  __shared__ float fs_lds[NP];
  __shared__ int   kt_lds[NP];
  __shared__ int   kf_lds[NP];

  const int g = blockIdx.x;
  const int b = g / NV;
  const int v = g - b * NV;
  const int p = threadIdx.x;                       // patch index 0..63

  // ---- load this thread's patch (12 contiguous floats, 16B aligned) ----
  const float* xp = x + ((size_t)(b * NP + p) * NV + v) * PL;
  float xv[PL];
  {
    const float4 q0 = ((const float4*)xp)[0];
    const float4 q1 = ((const float4*)xp)[1];
    const float4 q2 = ((const float4*)xp)[2];
    xv[0]=q0.x; xv[1]=q0.y; xv[2]=q0.z; xv[3]=q0.w;
    xv[4]=q1.x; xv[5]=q1.y; xv[6]=q1.z; xv[7]=q1.w;
    xv[8]=q2.x; xv[9]=q2.y; xv[10]=q2.z; xv[11]=q2.w;
  }
  #pragma unroll
  for (int i = 0; i < PL; ++i) x_lds[p][i] = xv[i];

  // ---- coefficient of variation (ddof=1) ----
  float sum = 0.f, sumsq = 0.f;
  #pragma unroll
  for (int i = 0; i < PL; ++i) { sum += xv[i]; sumsq += xv[i] * xv[i]; }
  const float mean = sum * (1.0f / PL);
  float var = (sumsq - sum * mean) * (1.0f / (PL - 1));
  var = var > 0.f ? var : 0.f;
  const float mycv = sqrtf(var) / (mean + 1e-6f);

  // ---- mean rFFT magnitude: twiddles are cos/sin(pi*m/6) exactly ----
  const float C12[12] = { 1.f,  0.8660254037844386f,  0.5f, 0.f, -0.5f, -0.8660254037844386f,
                         -1.f, -0.8660254037844386f, -0.5f, 0.f,  0.5f,  0.8660254037844386f};
  const float S12[12] = { 0.f,  0.5f,  0.8660254037844386f,  1.f,  0.8660254037844386f,  0.5f,
                          0.f, -0.5f, -0.8660254037844386f, -1.f, -0.8660254037844386f, -0.5f};
  float fsc = 0.f;
  #pragma unroll
  for (int k = 0; k < 7; ++k) {
    float re = 0.f, im = 0.f;
    #pragma unroll
    for (int n = 0; n < PL; ++n) {
      const int m = (k * n) % 12;
      re += xv[n] * C12[m];
      im -= xv[n] * S12[m];
    }
    fsc += sqrtf(re * re + im * im);
  }
  const float myfs = fsc * (1.0f / 7.0f);

  cv_lds[p] = mycv;
  fs_lds[p] = myfs;
  __syncthreads();

  // ---- stable top-k via rank counting (matches lax.top_k index tie-break) ----
  int rt = 0, rf = 0;
  #pragma unroll 8
  for (int q = 0; q < NP; ++q) {
    const float cq = cv_lds[q];
    const float fq = fs_lds[q];
    rt += (cq > mycv || (cq == mycv && q < p)) ? 1 : 0;  // time: largest cv masked
    rf += (fq < myfs || (fq == myfs && q < p)) ? 1 : 0;  // freq: smallest score masked
  }
  kt_lds[p] = (rt >= KT) ? 1 : 0;   // keep flag (not masked)
  kf_lds[p] = (rf >= KF) ? 1 : 0;
  __syncthreads();

  // ---- projection y = x@M + cvec via V_WMMA_F32_16X16X4_F32, fused epilogue ----
  const int lane  = threadIdx.x & 31;
  const int wid   = threadIdx.x >> 5;   // wave id (0,1)
  const int n     = lane & 15;          // output column (only n<12 stored)
  const int hi    = lane >> 4;          // lane half
  const int khalf = hi * 2;             // K sub-offset per A/B layout

  const float cw = ws[192 + n];         // cvec[n] (0 for pad cols)
  const float tv = ws[208 + n];
  const float fv = ws[224 + n];

  #pragma unroll
  for (int tile = 0; tile < 2; ++tile) {
    const int base = wid * 32 + tile * 16;
    const int row  = base + n;          // A-matrix row carried by this lane

    v8f acc;
    #pragma unroll
    for (int j = 0; j < 8; ++j) acc[j] = cw;   // C[m][n] = cvec[n]

    #pragma unroll
    for (int kc = 0; kc < PL; kc += 4) {
      v2f a, bm;
      a.x  = x_lds[row][kc + khalf];            // A: K={kc+khalf, kc+khalf+1}
      a.y  = x_lds[row][kc + khalf + 1];
      bm.x = ws[(kc + khalf) * 16 + n];         // B: M[K][n], padded cols are zero
      bm.y = ws[(kc + khalf + 1) * 16 + n];
      acc = __builtin_amdgcn_wmma_f32_16x16x4_f32(
          false, a, false, bm, (short)0, acc, false, false);
    }

    #pragma unroll
    for (int j = 0; j < 8; ++j) {
      const int tok = base + j + hi * 8;        // D row for VGPR j in this lane half
      const float y = acc[j];
      const float o = 0.5f * ((kt_lds[tok] ? y : tv) + (kf_lds[tok] ? y : fv));
      if (n < PL) {
        out[((size_t)(b * NP + tok) * NV + v) * PL + n] = o;
      }
    }
  }
}

extern "C" void kernel_launch(void* const* d_in, const int* in_sizes, int n_in,
                              void* d_out, int out_size, void* d_ws, size_t ws_size,
                              hipStream_t stream) {
  (void)in_sizes; (void)n_in; (void)out_size; (void)ws_size;
  const float* x     = (const float*)d_in[0];
  const float* W_in  = (const float*)d_in[1];
  const float* b_in  = (const float*)d_in[2];
  const float* W_out = (const float*)d_in[3];
  const float* b_out = (const float*)d_in[4];
  const float* tmt   = (const float*)d_in[5];
  const float* fmt   = (const float*)d_in[6];
  float* out = (float*)d_out;
  float* ws  = (float*)d_ws;

  tf_precompute<<<1, 256, 0, stream>>>(W_in, b_in, W_out, b_out, tmt, fmt, ws);
  tf_fused<<<dim3(BS * NV), dim3(64), 0, stream>>>(x, ws, out);
}